// ScaleSpecificSSM_80564996538673
// MI455X (gfx1250) — compile-verified
//
#include <hip/hip_runtime.h>
#include <hip/hip_bf16.h>
#include <stddef.h>

// ---------------------------------------------------------------------------
// ScaleSpecificSSM on MI455X (gfx1250, wave32)
//
//  y = LN( scan(x@B^T; s_t = s_{t-1}A^T + u_t) @ C^T + D*x )
//
// Pipeline (all fp32, WMMA f32_16x16x4 for the two big GEMMs):
//  K1 bx_gemm       : Bx[m,s]  = sum_h x[m,h] * B_w[s,h]   (TDM->LDS + WMMA)
//  K2 powers        : P[i] = (A^T)^(i+1), i = 0..63        (VALU, tiny)
//  K3 chunk_scan    : per-chunk local prefix states l_i    (VALU, 512-way)
//  K4 boundary_scan : serial carry over chunk boundaries   (VALU, tiny)
//  K5 output        : s = l + carry@P ; y = s@C^T + D*x ; LN (WMMA + LDS)
// ---------------------------------------------------------------------------

#define HIDDEN 1024
#define STATE  64
#define BATCH  8
#define SEQ    4096
#define ROWS   (BATCH * SEQ)     // 32768
#define CHUNK  64                // scan chunk length
#define NCHUNK (ROWS / CHUNK)    // 512 chunks total

typedef __attribute__((ext_vector_type(2))) float        v2f;
typedef __attribute__((ext_vector_type(8))) float        v8f;
typedef __attribute__((ext_vector_type(4))) unsigned int u32x4;
typedef __attribute__((ext_vector_type(8))) unsigned int u32x8;

// K1 LDS tile: 128 rows x 64 k, padded to stride 66 dwords (TDM pads for us)
#define K1_TROWS 128
#define K1_TK    64
#define K1_TSTR  66
#define K1_TILE_FLOATS (K1_TROWS * K1_TSTR)   // 8448 floats = 33792 B

// ---------------------------------------------------------------------------
// TDM: DMA one 2D tile (K1_TROWS x K1_TK fp32, row stride HIDDEN) from global
// into LDS at lds_off, inserting 2 pad dwords after every 64 dwords so the
// LDS image has stride 66 (bank-conflict-free for the WMMA A-frag reads).
// D# layout per CDNA5 ISA ch.8 (group0 128b, group1 256b; groups 2/3 = NULL).
// ---------------------------------------------------------------------------
__device__ __forceinline__ void tdm_load_tile(const float* gaddr, unsigned lds_off) {
  const unsigned long long ga = (unsigned long long)(uintptr_t)gaddr;
  u32x4 g0;
  g0[0] = 1u;                                   // count=1, user descriptor
  g0[1] = lds_off;                              // lds_addr (bytes)
  g0[2] = (unsigned)ga;                         // global_addr[31:0]
  g0[3] = (unsigned)((ga >> 32) & 0x01FFFFFFu)  // global_addr[56:32]
          | (2u << 30);                         // type = 2 ("image")
  u32x8 g1;
  g1[0] = (2u << 16)        // data_size = 2 -> 4 bytes
        | (1u << 20)        // pad_enable
        | (5u << 22)        // pad_interval code 5 -> every 64 dwords
        | (1u << 25);       // pad_amount  code 1 -> 2 dwords
  g1[1] = ((unsigned)HIDDEN & 0xFFFFu) << 16;   // tensor_dim0 = 1024 (low16)
  g1[2] = ((unsigned)ROWS   & 0xFFFFu) << 16;   // dim0 hi16=0 | tensor_dim1 lo16
  g1[3] = ((unsigned)K1_TK) << 16;              // dim1 hi16=0 | tile_dim0 = 64
  g1[4] = (unsigned)K1_TROWS;                   // tile_dim1 = 128, tile_dim2 = 0
  g1[5] = (unsigned)HIDDEN;                     // tensor_dim0_stride = 1024 (lo32)
  g1[6] = 0u;                                   // stride hi | dim1_stride lo (unused, 2D)
  g1[7] = 0u;
  asm volatile("tensor_load_to_lds %0, %1" :: "s"(g0), "s"(g1) : "memory");
}

// ---------------------------------------------------------------------------
// K1: Bx = x @ B_w^T   (M=32768, N=64, K=1024), fp32 WMMA 16x16x4.
// 8 waves / block, each wave owns a 16x64 output tile (4 accumulators).
// x tiles are DMA'd into LDS by the TDM, double-buffered on TENSORcnt.
// A-frag lane map: lane&15 -> row M, lane>>4 selects K pair {0,1}/{2,3}.
// B-frag lane map: lane&15 -> col N, same K-pair split (loads stay in L2).
// ---------------------------------------------------------------------------
__global__ __launch_bounds__(256)
void k1_bx_gemm(const float* __restrict__ x, const float* __restrict__ Bw,
                float* __restrict__ Bx) {
  extern __shared__ float xs[];                  // 2 * K1_TILE_FLOATS
  const int wave = threadIdx.x >> 5;
  const int lane = threadIdx.x & 31;
  const int l16  = lane & 15;
  const int hi   = lane >> 4;                    // 0/1 -> K base offset 0/2
  const int mblk = blockIdx.x * K1_TROWS;        // 128-row block tile
  const int m0   = mblk + wave * 16;             // this wave's 16-row tile

  const unsigned lds_base = (unsigned)(uintptr_t)&xs[0];

  v8f acc[4];
#pragma unroll
  for (int j = 0; j < 4; ++j)
#pragma unroll
    for (int r = 0; r < 8; ++r) acc[j][r] = 0.0f;

  // prime the pipeline: tile 0 -> buffer 0
  if (wave == 0)
    tdm_load_tile(x + (size_t)mblk * HIDDEN, lds_base);

  for (int t = 0; t < HIDDEN / K1_TK; ++t) {     // 16 K-tiles
    if (wave == 0) {
      if (t + 1 < HIDDEN / K1_TK) {
        // issue next tile into the other buffer, then wait for current
        tdm_load_tile(x + (size_t)mblk * HIDDEN + (t + 1) * K1_TK,
                      lds_base + ((t + 1) & 1) * (K1_TILE_FLOATS * 4));
        __builtin_amdgcn_s_wait_tensorcnt(1);
      } else {
        __builtin_amdgcn_s_wait_tensorcnt(0);
      }
    }
    __syncthreads();                             // publish tile to all waves

    const float* cur = xs + (t & 1) * K1_TILE_FLOATS;
    const int kt = t * K1_TK;
#pragma unroll
    for (int kk = 0; kk < K1_TK; kk += 4) {
      const int kb = kk + 2 * hi;
      v2f a = *(const v2f*)(cur + (wave * 16 + l16) * K1_TSTR + kb);
#pragma unroll
      for (int j = 0; j < 4; ++j) {
        const int n = j * 16 + l16;              // state index s
        v2f b = *(const v2f*)(Bw + (size_t)n * HIDDEN + kt + kb);
        acc[j] = __builtin_amdgcn_wmma_f32_16x16x4_f32(
            false, a, false, b, (short)0, acc[j], false, false);
      }
    }
    __syncthreads();                             // buffer may be reused
  }

  // C/D layout: VGPR r -> row (r + 8*hi), col = 16*j + l16
#pragma unroll
  for (int j = 0; j < 4; ++j)
#pragma unroll
    for (int r = 0; r < 8; ++r)
      Bx[(size_t)(m0 + r + 8 * hi) * STATE + j * 16 + l16] = acc[j][r];
}

// ---------------------------------------------------------------------------
// K2: matrix powers P[i] = (A^T)^(i+1), i = 0..63. One block, serial in i.
// Al padded to stride 66 to dodge LDS bank conflicts on column reads.
// ---------------------------------------------------------------------------
__global__ __launch_bounds__(512)
void k2_powers(const float* __restrict__ A, float* __restrict__ P) {
  __shared__ float Al[64 * 66];  // A, padded rows
  __shared__ float Pi[64 * 64];  // current power
  const int tid = threadIdx.x;

  for (int e = tid; e < 4096; e += 512) Al[(e >> 6) * 66 + (e & 63)] = A[e];
  __syncthreads();
  for (int e = tid; e < 4096; e += 512) {
    const int s = e >> 6, k = e & 63;
    const float v = Al[k * 66 + s];  // (A^T)[s][k] = A[k][s]
    Pi[e] = v;
    P[e]  = v;                       // slot 0 = (A^T)^1
  }
  __syncthreads();
  for (int p = 1; p < 64; ++p) {
    float nv[8];
#pragma unroll
    for (int q = 0; q < 8; ++q) {
      const int e = tid + q * 512;
      const int s = e >> 6, k = e & 63;
      float acc = 0.0f;
#pragma unroll 8
      for (int j = 0; j < 64; ++j) acc += Pi[s * 64 + j] * Al[k * 66 + j];
      nv[q] = acc;
    }
    __syncthreads();
#pragma unroll
    for (int q = 0; q < 8; ++q) {
      const int e = tid + q * 512;
      Pi[e] = nv[q];
      P[(size_t)p * 4096 + e] = nv[q];
    }
    __syncthreads();
  }
}

// ---------------------------------------------------------------------------
// K3: per-chunk local prefix scan. 512 blocks (one per chunk) x 64 threads.
// Thread k owns state element k; A row k kept in registers.
// Overwrites Bx in place with the local prefix states l_i.
// ---------------------------------------------------------------------------
__global__ __launch_bounds__(64)
void k3_chunk_scan(const float* __restrict__ A, float* __restrict__ BxL,
                   float* __restrict__ bl) {
  const int cb = blockIdx.x;   // chunk id = b*64 + c
  const int k  = threadIdx.x;  // state element
  __shared__ float s[64];
  float Arow[64];              // A[k][j] == (A^T)[j][k]
#pragma unroll 8
  for (int j = 0; j < 64; ++j) Arow[j] = A[k * 64 + j];
  s[k] = 0.0f;
  __syncthreads();

  const size_t base = (size_t)cb * CHUNK * STATE;
  float last = 0.0f;
  for (int i = 0; i < CHUNK; ++i) {
    float acc = BxL[base + i * STATE + k];       // u_i[k]
#pragma unroll 8
    for (int j = 0; j < 64; ++j) acc += s[j] * Arow[j];
    __syncthreads();                             // all reads of s done
    s[k] = acc;
    BxL[base + i * STATE + k] = acc;             // l_i[k]
    last = acc;
    __syncthreads();
  }
  bl[cb * STATE + k] = last;                     // l_{L-1}
}

// ---------------------------------------------------------------------------
// K4: serial scan over chunk boundaries (64 per batch, 8 batches in parallel).
// carryIn[b,c] = true state just before chunk c.
// carry' = carry @ (A^T)^64 + l_{L-1}(c)
// ---------------------------------------------------------------------------
__global__ __launch_bounds__(512)
void k4_boundary_scan(const float* __restrict__ P, const float* __restrict__ bl,
                      float* __restrict__ carryIn) {
  const int tid = threadIdx.x;
  const int b = tid >> 6, k = tid & 63;
  __shared__ float carry[BATCH][64];
  float col[64];
  const float* P64 = P + (size_t)63 * 4096;      // (A^T)^64
#pragma unroll 8
  for (int j = 0; j < 64; ++j) col[j] = P64[j * 64 + k];
  carry[b][k] = 0.0f;
  __syncthreads();
  for (int c = 0; c < NCHUNK / BATCH; ++c) {
    carryIn[(size_t)(b * 64 + c) * 64 + k] = carry[b][k];
    float acc = bl[(size_t)(b * 64 + c) * 64 + k];
#pragma unroll 8
    for (int j = 0; j < 64; ++j) acc += carry[b][j] * col[j];
    __syncthreads();
    carry[b][k] = acc;
    __syncthreads();
  }
}

// ---------------------------------------------------------------------------
// K5: fused fix-up + output GEMM + D*x + LayerNorm.
// One block per chunk (64 rows). Dynamic LDS: S_chunk(64x64,pad66) + Y(16x1024).
//  1) S[i,k] = l_i[k] + sum_s carry[s] * P_{i+1}[s,k]
//  2) per 16-row tile: Y = S_tile @ C^T via WMMA f32 16x16x4, += D*x
//  3) LayerNorm each row in LDS, write out.
// ---------------------------------------------------------------------------
#define SSTR 66  // padded stride of S_chunk in LDS

__global__ __launch_bounds__(256)
void k5_output(const float* __restrict__ x,  const float* __restrict__ Cw,
               const float* __restrict__ D,  const float* __restrict__ gamma,
               const float* __restrict__ beta,
               const float* __restrict__ L,       // local prefix states (Bx buf)
               const float* __restrict__ carryIn, // (NCHUNK,64)
               const float* __restrict__ P,       // powers
               float* __restrict__ out) {
  extern __shared__ float smem[];
  float* S = smem;                  // 64 * SSTR floats
  float* Y = smem + 64 * SSTR;      // 16 * 1024 floats
  __shared__ float carry[64];

  const int cb   = blockIdx.x;
  const int tid  = threadIdx.x;     // 256 threads = 8 waves
  const int wave = tid >> 5;
  const int lane = tid & 31;
  const int l16  = lane & 15;
  const int hi   = lane >> 4;
  const size_t rowbase = (size_t)cb * CHUNK;   // global (b*T + t) row base

  if (tid < 64) carry[tid] = carryIn[(size_t)cb * 64 + tid];
  __syncthreads();

  // ---- 1) fix-up: true states for this chunk ----
#pragma unroll
  for (int q = 0; q < 16; ++q) {
    const int e = tid + q * 256;
    const int i = e >> 6, k = e & 63;
    float acc = L[(rowbase + i) * STATE + k];
    const float* Pk = P + (size_t)i * 4096 + k;  // P_{i+1}[s][k], s-stride 64
#pragma unroll 8
    for (int s = 0; s < 64; ++s) acc += carry[s] * Pk[s * 64];
    S[i * SSTR + k] = acc;
  }
  __syncthreads();

  // ---- 2)+3) 16-row tiles: WMMA GEMM into LDS, then LayerNorm ----
  for (int mt = 0; mt < 4; ++mt) {
    // wave w covers n-tiles [8w, 8w+8): 1024 cols / 8 waves
#pragma unroll
    for (int ntq = 0; ntq < 8; ++ntq) {
      const int nt = wave * 8 + ntq;
      v8f acc;
#pragma unroll
      for (int r = 0; r < 8; ++r) acc[r] = 0.0f;
#pragma unroll
      for (int ks = 0; ks < 16; ++ks) {
        const int kb = ks * 4 + 2 * hi;
        v2f a = *(const v2f*)(S + (mt * 16 + l16) * SSTR + kb);
        // C^T[kb, h] = C_w[h, kb]
        v2f b = *(const v2f*)(Cw + (size_t)(nt * 16 + l16) * STATE + kb);
        acc = __builtin_amdgcn_wmma_f32_16x16x4_f32(
            false, a, false, b, (short)0, acc, false, false);
      }
#pragma unroll
      for (int r = 0; r < 8; ++r) {
        const int yrow = r + 8 * hi;             // 0..15 within tile
        const int h    = nt * 16 + l16;
        const size_t grow = rowbase + mt * 16 + yrow;
        Y[yrow * HIDDEN + h] = acc[r] + D[h] * x[grow * HIDDEN + h];
      }
    }
    __syncthreads();

    // LayerNorm: wave w handles rows 2w, 2w+1 of the 16-row tile
#pragma unroll
    for (int rr = 0; rr < 2; ++rr) {
      const int yrow = wave * 2 + rr;
      float sum = 0.0f, sq = 0.0f;
      for (int h = lane; h < HIDDEN; h += 32) {
        const float v = Y[yrow * HIDDEN + h];
        sum += v;
        sq  += v * v;
      }
#pragma unroll
      for (int off = 16; off > 0; off >>= 1) {
        sum += __shfl_xor(sum, off, 32);
        sq  += __shfl_xor(sq, off, 32);
      }
      const float mean = sum * (1.0f / HIDDEN);
      const float var  = sq * (1.0f / HIDDEN) - mean * mean;
      const float inv  = rsqrtf(var + 1e-5f);
      const size_t grow = rowbase + mt * 16 + yrow;
      for (int h = lane; h < HIDDEN; h += 32)
        out[grow * HIDDEN + h] =
            (Y[yrow * HIDDEN + h] - mean) * inv * gamma[h] + beta[h];
    }
    __syncthreads();
  }
}

// ---------------------------------------------------------------------------
// Host launcher. Inputs: x, A, B_w, C_w, D, gamma, beta (fp32).
// ---------------------------------------------------------------------------
extern "C" void kernel_launch(void* const* d_in, const int* in_sizes, int n_in,
                              void* d_out, int out_size, void* d_ws, size_t ws_size,
                              hipStream_t stream) {
  const float* x  = (const float*)d_in[0];
  const float* A  = (const float*)d_in[1];
  const float* Bw = (const float*)d_in[2];
  const float* Cw = (const float*)d_in[3];
  const float* D  = (const float*)d_in[4];
  const float* g  = (const float*)d_in[5];
  const float* be = (const float*)d_in[6];
  float* out = (float*)d_out;

  // workspace layout (floats)
  float* ws      = (float*)d_ws;
  float* Bx      = ws;                          // ROWS*STATE      = 2,097,152
  float* bl      = Bx + (size_t)ROWS * STATE;   // NCHUNK*STATE    =    32,768
  float* carryIn = bl + (size_t)NCHUNK * STATE; //                 =    32,768
  float* P       = carryIn + (size_t)NCHUNK * STATE; // 64*4096    =   262,144

  // K1: input GEMM, TDM double-buffered x tiles (256 blocks x 8 waves)
  const size_t smem1 = 2 * K1_TILE_FLOATS * sizeof(float);  // 67584 B
  k1_bx_gemm<<<ROWS / K1_TROWS, 256, smem1, stream>>>(x, Bw, Bx);
  // K2: matrix powers of A^T
  k2_powers<<<1, 512, 0, stream>>>(A, P);
  // K3: per-chunk local scans
  k3_chunk_scan<<<NCHUNK, 64, 0, stream>>>(A, Bx, bl);
  // K4: chunk-boundary serial scan
  k4_boundary_scan<<<1, 512, 0, stream>>>(P, bl, carryIn);
  // K5: fix-up + output GEMM + D*x + LayerNorm
  const size_t smem5 = (64 * SSTR + 16 * HIDDEN) * sizeof(float); // ~82 KB
  k5_output<<<NCHUNK, 256, smem5, stream>>>(x, Cw, D, g, be, Bx, carryIn, P, out);
}